// VAconv_68951404970015
// MI455X (gfx1250) — compile-verified
//
#include <hip/hip_runtime.h>
#include <hip/hip_bf16.h>

// ---------- WMMA vector types (gfx1250, wave32) ----------
typedef __attribute__((ext_vector_type(2))) float v2f;
typedef __attribute__((ext_vector_type(8))) float v8f;

#define D_FEAT 64

// ---------------------------------------------------------------------------
// Kernel 1: zero the neigh accumulator in workspace (must run every call:
// the scatter kernel accumulates with atomics).
// ---------------------------------------------------------------------------
__global__ void zero_f32(float* __restrict__ p, size_t n) {
    size_t i = (size_t)blockIdx.x * blockDim.x + threadIdx.x;
    size_t stride = (size_t)gridDim.x * blockDim.x;
    for (; i < n; i += stride) p[i] = 0.0f;
}

// ---------------------------------------------------------------------------
// Kernel 2: fused SDDMM + scatter-SpMM.
// One wave32 per edge. Lane l owns elements [2l, 2l+1] of the 64-wide row.
//   a_e = <feat[src], feat[dst]>   (wave reduction, 5 x shfl_xor on wave32)
//   neigh[dst] += feat[src] * a_e  (global_atomic_add_f32, L2-resident)
// ---------------------------------------------------------------------------
__global__ void sddmm_scatter(const float* __restrict__ feat,
                              const int* __restrict__ edge_src,
                              const int* __restrict__ edge_dst,
                              float* __restrict__ neigh,
                              int n_edges) {
    const int lane = threadIdx.x & 31;
    const int wave = threadIdx.x >> 5;
    const int e    = blockIdx.x * (blockDim.x >> 5) + wave;
    if (e >= n_edges) return;

    const int s = edge_src[e];
    const int d = edge_dst[e];

    const v2f hs = *(const v2f*)(feat + (size_t)s * D_FEAT + lane * 2);
    const v2f hd = *(const v2f*)(feat + (size_t)d * D_FEAT + lane * 2);

    // per-lane partial dot, then wave32 all-reduce
    float p = hs.x * hd.x + hs.y * hd.y;
    #pragma unroll
    for (int off = 16; off > 0; off >>= 1)
        p += __shfl_xor(p, off, 32);

    float* outp = neigh + (size_t)d * D_FEAT + lane * 2;
    atomicAdd(outp,     hs.x * p);
    atomicAdd(outp + 1, hs.y * p);
}

// ---------------------------------------------------------------------------
// Kernel 3: out[N,64] = neigh[N,64] @ W^T  via V_WMMA_F32_16X16X4_F32.
// One wave per 16-row M-tile; 4 v8f accumulators cover N=64; K in steps of 4.
//
// A-frag (16x4 f32, 2 VGPRs): lanes 0-15 hold rows M=0..15 with K={0,1},
//   lanes 16-31 hold the same rows with K={2,3}.
// B-frag (4x16 f32, 2 VGPRs), mirrored: lane supplies B[k][n] = W[n][k].
// D-frag (16x16 f32, 8 VGPRs): VGPR v = row (v + (lane>>4)*8), col = lane&15.
//
// Out-of-range rows: A row M only feeds D row M, so we simply CLAMP the load
// address (stay in-bounds, branchless, no EXEC churn) and never store those
// rows. For this problem N % 16 == 0, so the fast store path is always taken.
// ---------------------------------------------------------------------------
__global__ void gemm_wmma_f32(const float* __restrict__ neigh,
                              const float* __restrict__ W,   // [64,64] row-major [out][in]
                              float* __restrict__ out,
                              int n_nodes) {
    const int lane  = threadIdx.x & 31;
    const int wave  = threadIdx.x >> 5;
    const int tile  = blockIdx.x * (blockDim.x >> 5) + wave;
    const int row0  = tile * 16;
    if (row0 >= n_nodes) return;            // uniform per wave: EXEC stays all-ones

    const int l15   = lane & 15;
    const int khalf = (lane >> 4) * 2;      // 0 or 2

    int row = row0 + l15;
    if (row >= n_nodes) row = n_nodes - 1;  // clamp: keeps load in-bounds; row never stored

    const float* __restrict__ arow = neigh + (size_t)row * D_FEAT + khalf;
    const float* __restrict__ wrow = W + (size_t)l15 * D_FEAT + khalf;

    v8f acc0 = {}, acc1 = {}, acc2 = {}, acc3 = {};

    #pragma unroll
    for (int k0 = 0; k0 < D_FEAT; k0 += 4) {
        const v2f a  = *(const v2f*)(arow + k0);
        const v2f b0 = *(const v2f*)(wrow + 0 * 16 * D_FEAT + k0);
        const v2f b1 = *(const v2f*)(wrow + 1 * 16 * D_FEAT + k0);
        const v2f b2 = *(const v2f*)(wrow + 2 * 16 * D_FEAT + k0);
        const v2f b3 = *(const v2f*)(wrow + 3 * 16 * D_FEAT + k0);

        // 8 args: (neg_a, A, neg_b, B, c_mod, C, reuse_a, reuse_b)
        acc0 = __builtin_amdgcn_wmma_f32_16x16x4_f32(false, a, false, b0, (short)0, acc0, false, false);
        acc1 = __builtin_amdgcn_wmma_f32_16x16x4_f32(false, a, false, b1, (short)0, acc1, false, false);
        acc2 = __builtin_amdgcn_wmma_f32_16x16x4_f32(false, a, false, b2, (short)0, acc2, false, false);
        acc3 = __builtin_amdgcn_wmma_f32_16x16x4_f32(false, a, false, b3, (short)0, acc3, false, false);
    }

    // Store D: VGPR v -> row (row0 + v + (lane>>4)*8), col = nt*16 + (lane&15)
    const int mbase = row0 + (lane >> 4) * 8;
    float* __restrict__ obase = out + (size_t)mbase * D_FEAT + l15;

    if (row0 + 16 <= n_nodes) {
        // whole tile valid (always true when N % 16 == 0): unguarded stores
        #pragma unroll
        for (int v = 0; v < 8; ++v) {
            float* o = obase + (size_t)v * D_FEAT;
            o[0 * 16] = acc0[v];
            o[1 * 16] = acc1[v];
            o[2 * 16] = acc2[v];
            o[3 * 16] = acc3[v];
        }
    } else {
        #pragma unroll
        for (int v = 0; v < 8; ++v) {
            if (mbase + v < n_nodes) {
                float* o = obase + (size_t)v * D_FEAT;
                o[0 * 16] = acc0[v];
                o[1 * 16] = acc1[v];
                o[2 * 16] = acc2[v];
                o[3 * 16] = acc3[v];
            }
        }
    }
}

// ---------------------------------------------------------------------------
// Launch
// ---------------------------------------------------------------------------
extern "C" void kernel_launch(void* const* d_in, const int* in_sizes, int n_in,
                              void* d_out, int out_size, void* d_ws, size_t ws_size,
                              hipStream_t stream) {
    const float* feat     = (const float*)d_in[0];   // [N, 64]
    const int*   edge_src = (const int*)d_in[1];     // [E]
    const int*   edge_dst = (const int*)d_in[2];     // [E]
    const float* fc_w     = (const float*)d_in[3];   // [64, 64] row-major [out][in]
    float*       out      = (float*)d_out;           // [N, 64]

    const int n_nodes = in_sizes[0] / D_FEAT;
    const int n_edges = in_sizes[1];

    float* neigh = (float*)d_ws;                     // [N, 64] f32 accumulator
    const size_t neigh_elems = (size_t)n_nodes * D_FEAT;

    // 1) zero the accumulator (atomics below, must be fresh every call)
    zero_f32<<<2048, 256, 0, stream>>>(neigh, neigh_elems);

    // 2) fused SDDMM + scatter: 8 edges per 256-thread block (1 edge / wave32)
    {
        const int edges_per_block = 256 / 32;
        const int grid = (n_edges + edges_per_block - 1) / edges_per_block;
        sddmm_scatter<<<grid, 256, 0, stream>>>(feat, edge_src, edge_dst, neigh, n_edges);
    }

    // 3) WMMA f32 GEMM: 1 wave per 16-row tile, 8 waves per block
    {
        const int tiles = (n_nodes + 15) / 16;
        const int tiles_per_block = 256 / 32;
        const int grid = (tiles + tiles_per_block - 1) / tiles_per_block;
        gemm_wmma_f32<<<grid, 256, 0, stream>>>(neigh, fc_w, out, n_nodes);
    }
}